// WaveNetFF_76630806495640
// MI455X (gfx1250) — compile-verified
//
#include <hip/hip_runtime.h>

// CDNA5 / gfx1250 fused WaveNet-FF forward.
// One workgroup owns a 2048-sample chunk of one batch row, recomputes the
// 1200-sample receptive-field halo, keeps all 40 layers' activations in a
// double-buffered LDS tile, and runs each layer's 8x72 x 72xT conv-as-matmul
// on V_WMMA_F32_16X16X4_F32 (full-f32 matrix pipe => reference numerics).
// The 1x1 mix conv is accumulated into LDS layer-by-layer; activations never
// touch HBM. Per tile: 36 batched ds_loads -> one wait -> 18 WMMAs in two
// independent accumulation chains.

typedef __attribute__((ext_vector_type(2))) float v2f;
typedef __attribute__((ext_vector_type(8))) float v8f;

#define TT      262144          // time length
#define BATCH   8
#define CH      8
#define KK      9
#define LAYERS  40
#define CHUNK   2048            // output samples per workgroup
#define HALO    1216            // >= total receptive field (1200), mult of 16
#define TILE    (CHUNK + HALO)  // 3264 samples held in LDS
#define GUARD   64              // max in-layer look-back = (K-1)*dmax = 64
#define RS      (GUARD + TILE + 4) // row stride (floats), +4 to skew banks
#define NTHREADS 256
#define NWAVES  (NTHREADS / 32)
#define NTILES  (TILE / 16)     // 204 sixteen-wide WMMA time tiles
#define LW_OFF  (2 * CH * RS)       // LDS float offset: staged 8x72 weights
#define ACC_OFF (LW_OFF + CH * 72)  // LDS float offset: mix accumulator
#define SMEM_FLOATS (ACC_OFF + CHUNK)

#if __has_builtin(__builtin_amdgcn_global_load_async_to_lds_b32) && \
    __has_builtin(__builtin_amdgcn_s_wait_asynccnt)
#define HAVE_ASYNC_LDS 1
typedef __attribute__((address_space(1))) int gbl_i32;   // global (AS1) word
typedef __attribute__((address_space(3))) int lds_i32;   // LDS (AS3) word
#else
#define HAVE_ASYNC_LDS 0
#endif

__global__ __launch_bounds__(NTHREADS)
void wavenet_fused(const float* __restrict__ x,  const float* __restrict__ w0,
                   const float* __restrict__ ws, const float* __restrict__ bs,
                   const float* __restrict__ alphas,
                   const float* __restrict__ mix_w,
                   const float* __restrict__ mix_b,
                   float* __restrict__ out)
{
    extern __shared__ float smem[];
    float* lW   = smem + LW_OFF;
    float* accB = smem + ACC_OFF;

    const int tid  = threadIdx.x;
    const int lane = tid & 31;
    const int wave = tid >> 5;
    const int m    = lane & 15;        // WMMA row/col index within half-wave
    const bool hi  = lane >= 16;
    const int kb   = hi ? 2 : 0;       // K-pair base within 4-wide K chunk

    const int b         = blockIdx.y;
    const int tileStart = blockIdx.x * CHUNK;

    // --- zero both activation buffers (incl. guard zones) + mix accumulator.
    // (Guards stay zero forever => causal zero-padding; also kills any NaN
    //  garbage in unused channels for layer 0.)
    for (int j = tid; j < 2 * CH * RS; j += NTHREADS) smem[j] = 0.0f;
    for (int j = tid; j < CHUNK; j += NTHREADS) accB[j] = 0.0f;
    __syncthreads();   // LDS zeros complete before async loads may land

    // --- stage input x into channel 0 with left zero-pad
    {
        const float* xb = x + (size_t)b * TT;
#if HAVE_ASYNC_LDS
        for (int p = tid; p < TILE; p += NTHREADS) {
            int t = tileStart - HALO + p;
            if (t >= 0) {
                __builtin_amdgcn_global_load_async_to_lds_b32(
                    (gbl_i32*)(__UINTPTR_TYPE__)(xb + t),
                    (lds_i32*)(__UINTPTR_TYPE__)(smem + GUARD + p),
                    0, 0);
            }
        }
        __builtin_amdgcn_s_wait_asynccnt(0);
#else
        for (int p = tid; p < TILE; p += NTHREADS) {
            int t = tileStart - HALO + p;
            if (t >= 0) smem[GUARD + p] = xb[t];
        }
#endif
    }
    __syncthreads();

    float* bufIn  = smem;
    float* bufOut = smem + CH * RS;

    for (int layer = 0; layer < LAYERS; ++layer) {
        const int d = 1 << (layer & 3);   // dilation 1,2,4,8,1,2,...

        // --- stage this layer's weights as a zero-padded 8x72 matrix:
        //     lW[co*72 + ci*9 + k]  (layer 0: ci==0 only, rest zero)
        for (int j = tid; j < CH * 72; j += NTHREADS) {
            float wv;
            if (layer == 0) {
                int e = j % 72;
                wv = (e < KK) ? w0[(j / 72) * KK + e] : 0.0f;
            } else {
                wv = ws[(size_t)(layer - 1) * (CH * 72) + j];
            }
            lW[j] = wv;
        }
        if (layer + 1 < LAYERS)  // pre-pull next layer's weights into caches
            __builtin_prefetch(ws + (size_t)layer * (CH * 72), 0, 3);
        __syncthreads();

        // --- per-lane A fragments (weights). A is 16x4 per WMMA chunk:
        //     rows = c_out (rows 8..15 zero), K split lo/hi across half-waves.
        v2f afrag[18];
        #pragma unroll
        for (int g = 0; g < 18; ++g) {
            if (m < CH) {
                afrag[g] = *(const v2f*)(lW + m * 72 + 4 * g + kb); // ds_load_b64
            } else {
                afrag[g].x = 0.0f; afrag[g].y = 0.0f;
            }
        }

        // --- per-lane B gather offsets (loop-invariant across time tiles):
        //     element e = ci*9 + k reads h[ci][t - (8-k)*d]
        int off0[18], off1[18];
        #pragma unroll
        for (int g = 0; g < 18; ++g) {
            const int eL = 4 * g, eH = 4 * g + 2;
            int dL0 = ((eL    ) / 9) * RS - (8 - (eL    ) % 9) * d;
            int dL1 = ((eL + 1) / 9) * RS - (8 - (eL + 1) % 9) * d;
            int dH0 = ((eH    ) / 9) * RS - (8 - (eH    ) % 9) * d;
            int dH1 = ((eH + 1) / 9) * RS - (8 - (eH + 1) % 9) * d;
            off0[g] = hi ? dH0 : dL0;
            off1[g] = hi ? dH1 : dL1;
        }

        const float alpha = alphas[layer];
        float bias[8], mw8[8];
        #pragma unroll
        for (int j = 0; j < 8; ++j) {
            bias[j] = bs[layer * 8 + j];     // wave-uniform -> s_load
            mw8[j]  = mix_w[layer * 8 + j];
        }

        // --- hot loop: 16 time positions per WMMA tile, 18 K-chunks each.
        //     Batch all 36 DS loads first, then run two independent WMMA
        //     accumulation chains so the matrix pipe never waits per-chunk.
        for (int tt = wave; tt < NTILES; tt += NWAVES) {
            const int tBase = GUARD + tt * 16 + m;   // float index within row

            v2f bfrag[18];
            #pragma unroll
            for (int g = 0; g < 18; ++g) {
                bfrag[g].x = bufIn[tBase + off0[g]];  // ds_load_b32 (batched)
                bfrag[g].y = bufIn[tBase + off1[g]];
            }

            v8f acc0 = {}, acc1 = {};
            #pragma unroll
            for (int g = 0; g < 18; g += 2) {
                acc0 = __builtin_amdgcn_wmma_f32_16x16x4_f32(
                    false, afrag[g],     false, bfrag[g],     (short)0, acc0,
                    false, false);
                acc1 = __builtin_amdgcn_wmma_f32_16x16x4_f32(
                    false, afrag[g + 1], false, bfrag[g + 1], (short)0, acc1,
                    false, false);
            }

            // bias + PReLU + store + mix accumulation (rows live in lanes 0-15)
            if (lane < 16) {
                float mixsum = 0.0f;
                #pragma unroll
                for (int j = 0; j < 8; ++j) {
                    float y = acc0[j] + acc1[j] + bias[j];
                    y = (y >= 0.0f) ? y : alpha * y;
                    bufOut[j * RS + tBase] = y;
                    mixsum += mw8[j] * y;
                }
                const int t0 = tt * 16;
                if (t0 >= HALO) accB[t0 - HALO + m] += mixsum;
            }
        }
        __syncthreads();
        float* tmp = bufIn; bufIn = bufOut; bufOut = tmp;
    }

    // --- final mix bias + writeout of the valid CHUNK region
    const float mb = mix_b[0];
    float* outb = out + (size_t)b * TT + tileStart;
    for (int p = tid; p < CHUNK; p += NTHREADS) outb[p] = accB[p] + mb;
}

extern "C" void kernel_launch(void* const* d_in, const int* in_sizes, int n_in,
                              void* d_out, int out_size, void* d_ws, size_t ws_size,
                              hipStream_t stream) {
    const float* x    = (const float*)d_in[0];
    const float* w0   = (const float*)d_in[1];
    const float* wsp  = (const float*)d_in[2];
    const float* bs   = (const float*)d_in[3];
    const float* al   = (const float*)d_in[4];
    const float* mw   = (const float*)d_in[5];
    const float* mb   = (const float*)d_in[6];
    float* out = (float*)d_out;

    dim3 grid(TT / CHUNK, BATCH);           // 128 x 8 = 1024 workgroups
    size_t smem = SMEM_FLOATS * sizeof(float); // ~224 KB of the 320 KB WGP LDS
    wavenet_fused<<<grid, dim3(NTHREADS), smem, stream>>>(
        x, w0, wsp, bs, al, mw, mb, out);
}